// SAConv_57251914056256
// MI455X (gfx1250) — compile-verified
//
#include <hip/hip_runtime.h>
#include <hip/hip_bf16.h>

typedef float        v2f   __attribute__((ext_vector_type(2)));
typedef float        v8f   __attribute__((ext_vector_type(8)));
typedef unsigned int u32x4 __attribute__((ext_vector_type(4)));
typedef int          i32x4 __attribute__((ext_vector_type(4)));
typedef int          i32x8 __attribute__((ext_vector_type(8)));

constexpr int B_   = 8;
constexpr int C_   = 64;
constexpr int H_   = 56;
constexpr int W_   = 56;
constexpr int KW   = 7;
constexpr int PAD_ = 3;
constexpr int HP   = H_ + 2 * PAD_;      // 62
constexpr int WP   = W_ + 2 * PAD_;      // 62
constexpr int NPIX = HP * WP;            // 3844
constexpr int NTILE = (NPIX + 15) / 16;  // 241
constexpr int HWI  = H_ * W_;            // 3136

// ---------------------------------------------------------------------------
// v-projection: grouped 1x1 conv (8 groups of 8x8) over the zero-padded 62x62
// domain, + hard-swish. WMMA GEMM: block-diagonal 16x16 weight tile (2 groups)
// times a 16-pixel tile, fp32 accumulate via V_WMMA_F32_16X16X4_F32 (exact).
// All loads are unconditional (clamped address + 0/1 mask) so no per-lane
// saveexec branches; full EXEC is guaranteed at the WMMAs.
// ---------------------------------------------------------------------------
__global__ void vproj_wmma(const float* __restrict__ x,
                           const float* __restrict__ wv,
                           float* __restrict__ vfull)
{
    const int lane = threadIdx.x & 31;
    const int wid  = threadIdx.x >> 5;
    const int tile = blockIdx.x * 8 + wid;          // wave-uniform
    if (tile >= NTILE) return;                      // whole-wave exit
    const int c = blockIdx.y;                       // 16-channel chunk = groups 2c, 2c+1
    const int b = blockIdx.z;

    const int  nl = lane & 15;                      // N (pixel) for B/C, M (row) for A
    const bool hi = lane >= 16;
    const int  kb = hi ? 2 : 0;                     // K pair owned by this lane half

    const int  P  = tile * 16 + nl;                 // padded pixel id
    const int  py = P / WP, px = P % WP;
    const bool pvalid = P < NPIX;
    const int  iy = py - PAD_, ix = px - PAD_;
    const bool inter = pvalid && (unsigned)iy < (unsigned)H_ && (unsigned)ix < (unsigned)W_;
    const int  cy = min(max(iy, 0), H_ - 1);
    const int  cx = min(max(ix, 0), W_ - 1);
    const float xmask = inter ? 1.0f : 0.0f;

    // base pointers; the K index becomes a compile-time immediate offset
    const float* __restrict__ xb =
        x + ((size_t)(b * C_ + c * 16 + kb) * H_ + cy) * W_ + cx;
    const float maskLo = (nl < 8) ? 1.0f : 0.0f;    // K<8 block valid for rows 0..7
    const float maskHi = 1.0f - maskLo;             // K>=8 block valid for rows 8..15
    const float* __restrict__ wb =
        wv + (2 * c + (nl >= 8 ? 1 : 0)) * 64 + (nl & 7) * 8 + kb;

    v8f acc = {0.f, 0.f, 0.f, 0.f, 0.f, 0.f, 0.f, 0.f};

    // A 16x4: lanes0-15 (M=lane, K=vgpr),  lanes16-31 (M=lane-16, K=2+vgpr)
    // B 4x16: lanes0-15 (K=vgpr, N=lane),  lanes16-31 (K=2+vgpr, N=lane-16)
    #pragma unroll
    for (int s = 0; s < 4; ++s) {
        const float am = (s < 2) ? maskLo : maskHi;   // compile-time K<8 per step
        v2f a, bm;
        #pragma unroll
        for (int j = 0; j < 2; ++j) {
            a[j]  = am    * wb[(s & 1) * 4 + j];              // (k & 7) offset
            bm[j] = xmask * xb[(s * 4 + j) * HWI];            // channel-strided
        }
        acc = __builtin_amdgcn_wmma_f32_16x16x4_f32(false, a, false, bm,
                                                    (short)0, acc, false, false);
    }

    // C/D: VGPR r, lanes0-15 -> (M=r, N=lane), lanes16-31 -> (M=r+8, N=lane-16)
    if (pvalid) {
        const int mb = hi ? 8 : 0;
        float* __restrict__ vo =
            vfull + ((size_t)(b * C_ + c * 16 + mb) * HP + py) * WP + px;
        #pragma unroll
        for (int r = 0; r < 8; ++r) {
            float v = acc[r];
            float t = fminf(fmaxf(v + 3.0f, 0.0f), 6.0f);
            vo[r * (HP * WP)] = v * t * (1.0f / 6.0f);
        }
    }
}

// ---------------------------------------------------------------------------
// Fused 7x7 box mean. Each block: (batch b, 8-channel chunk, 8-row strip x full
// width). Halo tile (8ch x 14rows x 62cols, ~27KB) is staged into LDS by the
// Tensor Data Mover (3-D D#: dim0=62 contiguous, dim1=14 @stride 62, dim2=8
// @stride 3844, 4B elements), then the separable 7-tap sums run out of LDS.
// ---------------------------------------------------------------------------
__global__ void boxmean7_tdm(const float* __restrict__ vfull,
                             float* __restrict__ out)
{
    constexpr int CH   = 8;                 // channels per block
    constexpr int RT   = 8;                 // output rows per block
    constexpr int HR   = RT + KW - 1;       // 14 halo rows

    __shared__ float halo[CH * HR * WP];    // 8*14*62 f32 = 27776 B
    __shared__ float hrow[CH * HR * W_];    // 8*14*56 f32 = 25088 B

    const int y0  = blockIdx.x * RT;        // first output row (0..48)
    const int ch0 = blockIdx.y * CH;        // first channel
    const int b   = blockIdx.z;

    const float* __restrict__ gsrc =
        vfull + ((size_t)(b * C_ + ch0) * HP + y0) * WP;

#if __has_builtin(__builtin_amdgcn_tensor_load_to_lds)
    if (threadIdx.x < 32) {                 // one wave issues the DMA
        const unsigned long long ga = (unsigned long long)(uintptr_t)gsrc;
        const unsigned           lds = (unsigned)(uintptr_t)&halo[0];

        u32x4 g0;
        g0[0] = 1u;                                        // count=1 (valid user D#)
        g0[1] = lds;                                       // lds_addr
        g0[2] = (unsigned)ga;                              // global_addr[31:0]
        g0[3] = (unsigned)((ga >> 32) & 0x01FFFFFFu)       // global_addr[56:32]
              | (2u << 30);                                // type = 2 ("image")

        i32x8 g1;
        g1[0] = (2 << 16);                                 // data_size = 4B
        g1[1] = (WP << 16);                                // tensor_dim0 lo16 = 62
        g1[2] = (HR << 16);                                // dim0 hi=0 | tensor_dim1 lo16 = 14
        g1[3] = (WP << 16);                                // dim1 hi=0 | tile_dim0 = 62
        g1[4] = HR | (CH << 16);                           // tile_dim1 = 14, tile_dim2 = 8
        g1[5] = WP;                                        // tensor_dim0_stride = 62
        g1[6] = ((HP * WP) & 0xFFFF) << 16;                // stride0 hi=0 | dim1_stride lo16 = 3844
        g1[7] = (HP * WP) >> 16;                           // dim1_stride hi = 0

        i32x4 g2; g2[0] = CH; g2[1] = 0; g2[2] = 0; g2[3] = 0;   // tensor_dim2 = 8
        i32x4 g3; g3[0] = 0;  g3[1] = 0; g3[2] = 0; g3[3] = 0;

#if defined(__clang_major__) && (__clang_major__ >= 23)
        i32x8 g1b = {0, 0, 0, 0, 0, 0, 0, 0};
        __builtin_amdgcn_tensor_load_to_lds(g0, g1, g2, g3, g1b, 0);
#else
        __builtin_amdgcn_tensor_load_to_lds(g0, g1, g2, g3, 0);
#endif
        __builtin_amdgcn_s_wait_tensorcnt(0);
    }
#else
    // cooperative fallback: plain coalesced loads into LDS
    for (int i = threadIdx.x; i < CH * HR * WP; i += 256) {
        int cc  = i / (HR * WP);
        int rem = i - cc * (HR * WP);
        halo[i] = gsrc[(size_t)cc * (HP * WP) + rem];
    }
#endif
    __syncthreads();

    // stage A: 7-tap row sums (LDS -> LDS)
    for (int i = threadIdx.x; i < CH * HR * W_; i += 256) {
        int xo  = i % W_;
        int rr  = i / W_;                   // cc*HR + row
        const float* rp = &halo[rr * WP + xo];
        float s = 0.f;
        #pragma unroll
        for (int j = 0; j < KW; ++j) s += rp[j];
        hrow[i] = s;
    }
    __syncthreads();

    // stage B: 7-tap column sums * 1/49 -> out
    for (int i = threadIdx.x; i < CH * RT * W_; i += 256) {
        int xo  = i % W_;
        int t   = i / W_;
        int r   = t % RT;
        int cc  = t / RT;
        const float* cp = &hrow[(cc * HR + r) * W_ + xo];
        float s = 0.f;
        #pragma unroll
        for (int k = 0; k < KW; ++k) s += cp[k * W_];
        out[((size_t)(b * C_ + ch0 + cc) * H_ + (y0 + r)) * W_ + xo] =
            s * (1.0f / 49.0f);
    }
}

extern "C" void kernel_launch(void* const* d_in, const int* in_sizes, int n_in,
                              void* d_out, int out_size, void* d_ws, size_t ws_size,
                              hipStream_t stream)
{
    // setup_inputs order: x, wq, wk, wv, rel_h, rel_w (all fp32).
    // scale = sqrt(8 // 64) == 0 -> softmax uniform -> q/k/rel are dead inputs;
    // reference reduces to a 7x7 box mean of hswish(grouped_1x1(pad(x), wv)).
    const float* x  = (const float*)d_in[0];
    const float* wv = (const float*)d_in[3];
    float* out   = (float*)d_out;
    float* vfull = (float*)d_ws;            // 8*64*62*62 f32 (~7.9 MB)

    dim3 gp((NTILE + 7) / 8, C_ / 16, B_);  // (31, 4, 8), 8 waves/block
    vproj_wmma<<<gp, 256, 0, stream>>>(x, wv, vfull);

    dim3 gb(H_ / 8, C_ / 8, B_);            // (7, 8, 8)
    boxmean7_tdm<<<gb, 256, 0, stream>>>(vfull, out);
}